// GATEncoder_84069689852147
// MI455X (gfx1250) — compile-verified
//
#include <hip/hip_runtime.h>
#include <hip/hip_bf16.h>

#define GAT_N      10000
#define GAT_E      160000
#define GAT_HEADS  32
#define GAT_INDIM  128
#define GAT_H1     64
#define GAT_H2     32

typedef __attribute__((ext_vector_type(2))) float v2f;
typedef __attribute__((ext_vector_type(8))) float v8f;

// ---------------------------------------------------------------------------
// WMMA f32 GEMM:  C[M x ldb] = A[M x K] * B[K x ldb]
// One wave per 16-row strip x NT 16-col tiles. M % 16 == 0, ldb % (16*NT) == 0.
// A fragments (16x4 f32 tiles) held in registers across the NT column tiles.
// ---------------------------------------------------------------------------
template <int K, int NT>
__global__ __launch_bounds__(32)
void gemm_wmma_f32(const float* __restrict__ A, const float* __restrict__ B,
                   float* __restrict__ C, int ldb) {
  const int lane = threadIdx.x & 31;
  const int half = lane >> 4;   // 0: K+0/1, 1: K+2/3 (32-bit A/B VGPR layout)
  const int l    = lane & 15;
  const int m0 = blockIdx.x * 16;
  const int n0 = blockIdx.y * (16 * NT);

  v2f a[K / 4];
#pragma unroll
  for (int k0 = 0; k0 < K / 4; ++k0) {
    const float* ap = A + (size_t)(m0 + l) * K + 4 * k0 + 2 * half;
    v2f av; av.x = ap[0]; av.y = ap[1];
    a[k0] = av;
  }

  v8f acc[NT];
#pragma unroll
  for (int t = 0; t < NT; ++t)
#pragma unroll
    for (int v = 0; v < 8; ++v) acc[t][v] = 0.0f;

#pragma unroll
  for (int t = 0; t < NT; ++t) {
    const int n = n0 + 16 * t + l;
#pragma unroll
    for (int k0 = 0; k0 < K / 4; ++k0) {
      const int kk = 4 * k0 + 2 * half;
      v2f bv; bv.x = B[(size_t)kk * ldb + n]; bv.y = B[(size_t)(kk + 1) * ldb + n];
      acc[t] = __builtin_amdgcn_wmma_f32_16x16x4_f32(
          false, a[k0], false, bv, (short)0, acc[t], false, false);
    }
  }

#pragma unroll
  for (int t = 0; t < NT; ++t) {
    const int n = n0 + 16 * t + l;
#pragma unroll
    for (int v = 0; v < 8; ++v)
      C[(size_t)(m0 + v + 8 * half) * ldb + n] = acc[t][v];
  }
}

// ---------------------------------------------------------------------------
// Per-(node,head) attention logits: aS[n,h] = dot(xp[n,h,:], a_src[h,:]) etc.
// tid == n*32+h matches the [N,H,C] layout of xp.
// ---------------------------------------------------------------------------
template <int C>
__global__ void alpha_kernel(const float* __restrict__ xp,
                             const float* __restrict__ a_src,
                             const float* __restrict__ a_dst,
                             float* __restrict__ aS, float* __restrict__ aD,
                             int n_nodes) {
  int tid = blockIdx.x * blockDim.x + threadIdx.x;
  if (tid >= n_nodes * GAT_HEADS) return;
  int h = tid & (GAT_HEADS - 1);
  const float* xv = xp + (size_t)tid * C;
  const float* as = a_src + h * C;
  const float* ad = a_dst + h * C;
  float s = 0.f, d = 0.f;
#pragma unroll
  for (int c = 0; c < C; ++c) { float x = xv[c]; s += x * as[c]; d += x * ad[c]; }
  aS[tid] = s;
  aD[tid] = d;
}

// --------------------------- CSR construction ------------------------------
__global__ void fill_i32(int* __restrict__ p, int v, int n) {
  int i = blockIdx.x * blockDim.x + threadIdx.x;
  if (i < n) p[i] = v;
}

__global__ void copy_i32(const int* __restrict__ a, int* __restrict__ b, int n) {
  int i = blockIdx.x * blockDim.x + threadIdx.x;
  if (i < n) b[i] = a[i];
}

__global__ void count_deg(const int* __restrict__ dst, int* __restrict__ deg, int e) {
  int i = blockIdx.x * blockDim.x + threadIdx.x;
  if (i < e) atomicAdd(&deg[dst[i]], 1);
}

// Single-block exclusive scan: rowstart[0..n], rowstart[n] = total.
__global__ void scan_kernel(const int* __restrict__ deg, int* __restrict__ rowstart, int n) {
  __shared__ int part[1024];
  const int t = threadIdx.x;
  const int CH = (n + 1023) >> 10;
  const int base = t * CH;
  int s = 0;
  for (int i = 0; i < CH; ++i) { int idx = base + i; if (idx < n) s += deg[idx]; }
  part[t] = s;
  __syncthreads();
  for (int off = 1; off < 1024; off <<= 1) {
    int v = (t >= off) ? part[t - off] : 0;
    __syncthreads();
    part[t] += v;
    __syncthreads();
  }
  int run = (t == 0) ? 0 : part[t - 1];
  for (int i = 0; i < CH; ++i) {
    int idx = base + i;
    if (idx < n) { rowstart[idx] = run; run += deg[idx]; }
  }
  if (t == 0) rowstart[n] = part[1023];
}

__global__ void scatter_edges(const int* __restrict__ src, const int* __restrict__ dst,
                              int* __restrict__ cursor, int* __restrict__ esrc, int e) {
  int i = blockIdx.x * blockDim.x + threadIdx.x;
  if (i < e) {
    int p = atomicAdd(&cursor[dst[i]], 1);
    esrc[p] = src[i];
  }
}

// ---------------------------------------------------------------------------
// Per-node aggregation: one wave per destination node, lane == head (H == 32).
// Pass 1: segment max; pass 2: exp-sum; pass 3: weighted channel accumulate.
// Head-mean done via butterfly reduce-scatter (static register indices only),
// leaving every lane with acc[j] (j < C/32) = sum over heads h of head h's channel C/32*h + j; then bias + ELU.
// ---------------------------------------------------------------------------
template <int C>
__global__ __launch_bounds__(32)
void aggregate_kernel(const int* __restrict__ rowstart, const int* __restrict__ esrc,
                      const float* __restrict__ aS, const float* __restrict__ aD,
                      const float* __restrict__ xp, const float* __restrict__ bias,
                      float* __restrict__ z, float* __restrict__ outc, int outoff) {
  const int n = blockIdx.x;
  const int lane = threadIdx.x;
  const int row = rowstart[n];
  const int end = rowstart[n + 1];
  const float ad = aD[n * GAT_HEADS + lane];

  float m = -3.402823466e38f;
  for (int i = row; i < end; ++i) {
    int s = esrc[i];
    float t = aS[s * GAT_HEADS + lane] + ad;
    t = t > 0.f ? t : 0.2f * t;
    m = fmaxf(m, t);
  }
  float ssum = 0.f;
  for (int i = row; i < end; ++i) {
    int s = esrc[i];
    float t = aS[s * GAT_HEADS + lane] + ad;
    t = t > 0.f ? t : 0.2f * t;
    ssum += __expf(t - m);
  }
  const float inv = 1.f / (ssum + 1e-16f);

  float acc[C];
#pragma unroll
  for (int c = 0; c < C; ++c) acc[c] = 0.f;

  for (int i = row; i < end; ++i) {
    int s = esrc[i];
    float t = aS[s * GAT_HEADS + lane] + ad;
    t = t > 0.f ? t : 0.2f * t;
    float w = __expf(t - m) * inv;
    const float4* xv = (const float4*)(xp + ((size_t)s * GAT_HEADS + lane) * C);
#pragma unroll
    for (int c4 = 0; c4 < C / 4; ++c4) {
      float4 v = xv[c4];
      acc[4 * c4 + 0] += w * v.x;
      acc[4 * c4 + 1] += w * v.y;
      acc[4 * c4 + 2] += w * v.z;
      acc[4 * c4 + 3] += w * v.w;
    }
  }

  // Butterfly reduce-scatter over 32 lanes (heads). After 5 steps, every lane
  // holds acc[j] = sum over heads h of head h's channel (C/32)*h + j, j < C/32.
  int L = C;
#pragma unroll
  for (int off = 16; off > 0; off >>= 1) {
    const int half = L / 2;
    const bool upper = (lane & off) != 0;
#pragma unroll
    for (int i = 0; i < C; ++i) {   // only first `half` entries meaningful
      if (i < half) {
        float kv = upper ? acc[i + half] : acc[i];
        float other = __shfl_xor(kv, off, 32);
        acc[i] = kv + other;
      }
    }
    L = half;
    if (L == 1 && off > 1) {
      // never taken for C == 32 or 64 (L > 1 while off > 1).
    }
  }

  const float invH = 1.0f / (float)GAT_HEADS;
  const int CPL = C / 32;           // channels per lane after reduce-scatter
#pragma unroll
  for (int j = 0; j < CPL; ++j) {
    int c = CPL * lane + j;
    float zz = acc[j] * invH + bias[c];
    zz = zz > 0.f ? zz : (__expf(zz) - 1.f);   // ELU(alpha=1)
    if (z) z[(size_t)n * C + c] = zz;
    outc[(size_t)n * 96 + outoff + c] = zz;
  }
}

// ---------------------------------------------------------------------------
extern "C" void kernel_launch(void* const* d_in, const int* in_sizes, int n_in,
                              void* d_out, int out_size, void* d_ws, size_t ws_size,
                              hipStream_t stream) {
  (void)in_sizes; (void)n_in; (void)out_size; (void)ws_size;

  const float* x     = (const float*)d_in[0];
  const int*   ei    = (const int*)d_in[1];       // [2, E]
  const float* W1    = (const float*)d_in[2];     // [128, 2048]
  const float* aS1w  = (const float*)d_in[3];     // [32, 64]
  const float* aD1w  = (const float*)d_in[4];
  const float* b1    = (const float*)d_in[5];     // [64]
  const float* W2    = (const float*)d_in[6];     // [64, 1024]
  const float* aS2w  = (const float*)d_in[7];     // [32, 32]
  const float* aD2w  = (const float*)d_in[8];
  const float* b2    = (const float*)d_in[9];     // [32]
  float*       out   = (float*)d_out;             // [N, 96]

  const int* src = ei;
  const int* dst = ei + GAT_E;

  // --- workspace layout (256B aligned) ---
  char* base = (char*)d_ws;
  size_t off = 0;
  auto alloc = [&](size_t bytes) -> char* {
    char* p = base + off;
    off = (off + bytes + 255) & ~(size_t)255;
    return p;
  };
  float* xp1  = (float*)alloc((size_t)GAT_N * GAT_HEADS * GAT_H1 * 4);  // 81.9 MB
  float* xp2  = (float*)alloc((size_t)GAT_N * GAT_HEADS * GAT_H2 * 4);  // 41.0 MB
  float* aS1  = (float*)alloc((size_t)GAT_N * GAT_HEADS * 4);
  float* aD1  = (float*)alloc((size_t)GAT_N * GAT_HEADS * 4);
  float* aS2  = (float*)alloc((size_t)GAT_N * GAT_HEADS * 4);
  float* aD2  = (float*)alloc((size_t)GAT_N * GAT_HEADS * 4);
  float* z1   = (float*)alloc((size_t)GAT_N * GAT_H1 * 4);
  int*   deg      = (int*)alloc((size_t)GAT_N * 4);
  int*   rowstart = (int*)alloc((size_t)(GAT_N + 1) * 4);
  int*   cursor   = (int*)alloc((size_t)GAT_N * 4);
  int*   esrc     = (int*)alloc((size_t)GAT_E * 4);

  const int TB = 256;
  const int gE = (GAT_E + TB - 1) / TB;
  const int gN = (GAT_N + TB - 1) / TB;
  const int gNH = (GAT_N * GAT_HEADS + TB - 1) / TB;

  // --- CSR build (shared by both layers) ---
  fill_i32<<<gN, TB, 0, stream>>>(deg, 0, GAT_N);
  count_deg<<<gE, TB, 0, stream>>>(dst, deg, GAT_E);
  scan_kernel<<<1, 1024, 0, stream>>>(deg, rowstart, GAT_N);
  copy_i32<<<gN, TB, 0, stream>>>(rowstart, cursor, GAT_N);
  scatter_edges<<<gE, TB, 0, stream>>>(src, dst, cursor, esrc, GAT_E);

  // --- Layer 1 ---
  {
    dim3 grid(GAT_N / 16, (GAT_HEADS * GAT_H1) / (16 * 4));  // (625, 32)
    gemm_wmma_f32<GAT_INDIM, 4><<<grid, 32, 0, stream>>>(x, W1, xp1, GAT_HEADS * GAT_H1);
  }
  alpha_kernel<GAT_H1><<<gNH, TB, 0, stream>>>(xp1, aS1w, aD1w, aS1, aD1, GAT_N);
  aggregate_kernel<GAT_H1><<<GAT_N, 32, 0, stream>>>(rowstart, esrc, aS1, aD1,
                                                     xp1, b1, z1, out, 0);

  // --- Layer 2 ---
  {
    dim3 grid(GAT_N / 16, (GAT_HEADS * GAT_H2) / (16 * 4));  // (625, 16)
    gemm_wmma_f32<GAT_H1, 4><<<grid, 32, 0, stream>>>(z1, W2, xp2, GAT_HEADS * GAT_H2);
  }
  alpha_kernel<GAT_H2><<<gNH, TB, 0, stream>>>(xp2, aS2w, aD2w, aS2, aD2, GAT_N);
  aggregate_kernel<GAT_H2><<<GAT_N, 32, 0, stream>>>(rowstart, esrc, aS2, aD2,
                                                     xp2, b2, nullptr, out, GAT_H1);
}